// Weighted3DTransformNormLoss_37933151159024
// MI455X (gfx1250) — compile-verified
//
#include <hip/hip_runtime.h>

// Weighted3DTransformNormLoss for gfx1250 (MI455X, wave32, WMMA).
// B=16, K=8, H=240, W=320, EPS=0.002, NORM_WT=0.5, NORM_PER_PT=False.
//
// Per 16-pixel tile: Out[c, pix] = sum_k Tfm[k][c] * mask[k, pix]  (c = i*4+j,
// 12 used of 16) is computed as two V_WMMA_F32_16X16X4_F32 (K=8 in two K=4
// chunks).  D layout puts rows i=0,1 in lanes 0-15 and row i=2 in lanes 16-31;
// since the result is a global scalar sum, the halves never need to exchange.

typedef __attribute__((ext_vector_type(2))) float v2f;
typedef __attribute__((ext_vector_type(8))) float v8f;

#define HW          76800            // 240*320
#define KK          8
#define BB          16
#define TILES_PER_B (HW / 16)        // 4800 tiles of 16 pixels
#define NUM_TILES   (BB * TILES_PER_B)
#define GRID        512
#define BLOCK       256
#define WAVES       (GRID * (BLOCK / 32))

__global__ __launch_bounds__(BLOCK) void w3d_loss_kernel(
    const float* __restrict__ pts,    // [B,3,H,W]
    const float* __restrict__ masks,  // [B,K,H,W]
    const float* __restrict__ tfms,   // [B,K,3,4]
    const float* __restrict__ tflow,  // [B,3,H,W]
    const float* __restrict__ numpts, // [B]
    float* __restrict__ partial)      // [GRID]
{
  const int lane  = threadIdx.x & 31;
  const int n     = lane & 15;      // pixel-in-tile / matrix column
  const int half  = lane >> 4;      // K-half select (per WMMA A/B layout)
  const int waveId = blockIdx.x * (BLOCK / 32) + (threadIdx.x >> 5);

  // Hoist the per-batch scale 0.5/(numpts[b]*B): one division per batch per
  // block instead of an IEEE div chain every loop iteration.
  __shared__ float invs[BB];
  if (threadIdx.x < BB) invs[threadIdx.x] = 0.5f / (numpts[threadIdx.x] * (float)BB);
  __syncthreads();

  const float selB = (half == 0) ? 1.0f : 0.0f;  // mask for second error row

  float acc = 0.0f;

  // Wave-uniform trip count -> EXEC is all-ones at every WMMA.
  for (int t = waveId; t < NUM_TILES; t += WAVES) {
    const int b   = t / TILES_PER_B;
    const int pix = (t - b * TILES_PER_B) * 16 + n;   // 64B-aligned tile base

    const size_t mbase = (size_t)b * KK * HW;
    const size_t pbase = (size_t)b * 3 * HW;

    // ---- A = Tfm^T : A[m = coeff c (0..11, pad 16), k = mask index] ----
    // lane holds A[m=n, k=2*half+{0,1}] for each K-chunk (k base 0 and 4).
    const int m = n;
    const float* Tb = tfms + (size_t)b * KK * 12;     // Tfm[k][c], c = i*4+j
    v2f a0, a1;
    if (m < 12) {
      a0.x = Tb[(0 + 2 * half + 0) * 12 + m];
      a0.y = Tb[(0 + 2 * half + 1) * 12 + m];
      a1.x = Tb[(4 + 2 * half + 0) * 12 + m];
      a1.y = Tb[(4 + 2 * half + 1) * 12 + m];
    } else {
      a0.x = 0.0f; a0.y = 0.0f; a1.x = 0.0f; a1.y = 0.0f;
    }

    // ---- B = masks : B[k, n = pixel]; lane holds B[2*half+{0,1}, n] ----
    v2f b0, b1;
    b0.x = masks[mbase + (size_t)(0 + 2 * half + 0) * HW + pix];
    b0.y = masks[mbase + (size_t)(0 + 2 * half + 1) * HW + pix];
    b1.x = masks[mbase + (size_t)(4 + 2 * half + 0) * HW + pix];
    b1.y = masks[mbase + (size_t)(4 + 2 * half + 1) * HW + pix];

    // Prefetch next tile's first mask line (streaming hint).
    {
      const int t2 = t + WAVES;
      if (t2 < NUM_TILES) {
        const int b2   = t2 / TILES_PER_B;
        const int pix2 = (t2 - b2 * TILES_PER_B) * 16 + n;
        __builtin_prefetch(&masks[(size_t)b2 * KK * HW + (size_t)(2 * half) * HW + pix2], 0, 1);
      }
    }

    // ---- D = A x B (+C):  Out[c, pixel] = sum_k Tfm[k][c] * mask_k ----
    v8f c = {};
    c = __builtin_amdgcn_wmma_f32_16x16x4_f32(false, a0, false, b0, (short)0, c, false, false);
    c = __builtin_amdgcn_wmma_f32_16x16x4_f32(false, a1, false, b1, (short)0, c, false, false);
    // half==0 lane n: Out[c=0..7] (rows i=0,1);  half==1: Out[c=8..11] (i=2, pad 0).

    const float px = pts[pbase + (size_t)0 * HW + pix];
    const float py = pts[pbase + (size_t)1 * HW + pix];
    const float pz = pts[pbase + (size_t)2 * HW + pix];

    // Branchless: both halves evaluate row A (i=0 or i=2) from c[0..3];
    // row B (i=1, from c[4..7]) is zero-masked on half==1, where c[4..7] is
    // the WMMA zero padding (rows 12..15) -> eB finite, selB kills it.
    const float tfA = tflow[pbase + (size_t)(half ? 2 : 0) * HW + pix];
    const float tfB = tflow[pbase + (size_t)1 * HW + pix];  // same lines as half-0
    const float subA = half ? pz : px;

    const float eA = c[0] * px + c[1] * py + c[2] * pz + c[3] - subA - tfA;
    const float eB = c[4] * px + c[5] * py + c[6] * pz + c[7] - py - tfB;
    const float sA = fmaxf(0.5f * tfA * tfA, 0.002f);
    const float sB = fmaxf(0.5f * tfB * tfB, 0.002f);

    const float contrib = (eA * eA) / sA + selB * ((eB * eB) / sB);
    acc += contrib * invs[b];
  }

  // Deterministic block reduction.
  __shared__ float red[BLOCK];
  red[threadIdx.x] = acc;
  __syncthreads();
  for (int s = BLOCK / 2; s > 0; s >>= 1) {
    if (threadIdx.x < s) red[threadIdx.x] += red[threadIdx.x + s];
    __syncthreads();
  }
  if (threadIdx.x == 0) partial[blockIdx.x] = red[0];
}

__global__ __launch_bounds__(BLOCK) void w3d_reduce_kernel(
    const float* __restrict__ partial, float* __restrict__ out)
{
  __shared__ float red[BLOCK];
  float v = 0.0f;
  for (int i = threadIdx.x; i < GRID; i += BLOCK) v += partial[i];
  red[threadIdx.x] = v;
  __syncthreads();
  for (int s = BLOCK / 2; s > 0; s >>= 1) {
    if (threadIdx.x < s) red[threadIdx.x] += red[threadIdx.x + s];
    __syncthreads();
  }
  if (threadIdx.x == 0) out[0] = red[0];
}

extern "C" void kernel_launch(void* const* d_in, const int* in_sizes, int n_in,
                              void* d_out, int out_size, void* d_ws, size_t ws_size,
                              hipStream_t stream) {
  (void)in_sizes; (void)n_in; (void)out_size; (void)ws_size;
  const float* pts    = (const float*)d_in[0]; // inputpts    [B,3,H,W]
  const float* masks  = (const float*)d_in[1]; // inputmasks  [B,K,H,W]
  const float* tfms   = (const float*)d_in[2]; // inputtfms   [B,K,3,4]
  const float* tflow  = (const float*)d_in[3]; // targetflows [B,3,H,W]
  const float* numpts = (const float*)d_in[4]; // numpts      [B]
  float* partial = (float*)d_ws;               // GRID floats of scratch
  float* out     = (float*)d_out;              // scalar f32

  w3d_loss_kernel<<<GRID, BLOCK, 0, stream>>>(pts, masks, tfms, tflow, numpts, partial);
  w3d_reduce_kernel<<<1, BLOCK, 0, stream>>>(partial, out);
}